// CharacterModel_57939108823307
// MI455X (gfx1250) — compile-verified
//
#include <hip/hip_runtime.h>
#include <hip/hip_bf16.h>

#define GH 16   // hidden size
#define GL 15   // sequence length

typedef __attribute__((ext_vector_type(2))) float v2f;
typedef __attribute__((ext_vector_type(4))) float v4f;
typedef __attribute__((ext_vector_type(8))) float v8f;

__device__ __forceinline__ v2f mkv2(float a, float b) { v2f r; r.x = a; r.y = b; return r; }

// D(16x16 f32) += A(16x4 f32) * B(4x16 f32), full f32 precision on the matrix pipe
__device__ __forceinline__ v8f wmma_k4(v2f a, v2f b, v8f c) {
  return __builtin_amdgcn_wmma_f32_16x16x4_f32(false, a, false, b, (short)0, c, false, false);
}

__device__ __forceinline__ float fsigmoid(float x) { return 1.0f / (1.0f + __expf(-x)); }
__device__ __forceinline__ float ftanh_(float x) {
  float e = __expf(-2.0f * x);
  return (1.0f - e) / (1.0f + e);
}

__global__ __launch_bounds__(32) void bigru_ln_kernel(
    const int* __restrict__ ids, const float* __restrict__ emb,
    const float* __restrict__ w_ih_f, const float* __restrict__ w_hh_f,
    const float* __restrict__ b_ih_f, const float* __restrict__ b_hh_f,
    const float* __restrict__ w_ih_b, const float* __restrict__ w_hh_b,
    const float* __restrict__ b_ih_b, const float* __restrict__ b_hh_b,
    const float* __restrict__ gamma, const float* __restrict__ beta,
    float* __restrict__ out, int n_seq)
{
  // forward outputs [t][hid][s], embedding table, store stage  (~17.5 KB total)
  __shared__ float flds[GL * 16 * 16];
  __shared__ float elds[5 * 16];
  __shared__ __align__(16) float ystage[16 * 32];

  const int  lane = threadIdx.x;       // 0..31 (wave32)
  const int  m16  = lane & 15;
  const bool hi   = lane >= 16;
  const long nbase = (long)blockIdx.x * 16;

  // stage embedding (5x16) into LDS
  for (int i = lane; i < 80; i += 32) elds[i] = emb[i];

  // per-lane: pack the 15 bucket codes (3 bits each) of sequence s = lane&15.
  // Both lane halves compute the same sequence's buckets (both need them).
  unsigned bp0 = 0, bp1 = 0;
  {
    long n = nbase + m16;
    bool ok = n < (long)n_seq;
    const int* idp = ids + n * GL;
#pragma unroll
    for (int t = 0; t < GL; ++t) {
      int id = ok ? idp[t] : 0;
      unsigned bk = (unsigned)((id >= 1) + (id >= 11) + (id >= 37) + (id >= 63));
      if (t < 10) bp0 |= bk << (3 * t);
      else        bp1 |= bk << (3 * (t - 10));
    }
  }

  // per-lane layernorm affine params for this lane's channels
  float gamF[8], betF[8], gamB[8], betB[8];
#pragma unroll
  for (int r = 0; r < 8; ++r) {
    int cf = r + (hi ? 8 : 0);
    gamF[r] = gamma[cf];       betF[r] = beta[cf];
    gamB[r] = gamma[cf + 16];  betB[r] = beta[cf + 16];
  }

  const v8f vzero = {0.f, 0.f, 0.f, 0.f, 0.f, 0.f, 0.f, 0.f};

  for (int dir = 0; dir < 2; ++dir) {
    const float* Wih = dir ? w_ih_b : w_ih_f;
    const float* Whh = dir ? w_hh_b : w_hh_f;
    const float* Bih = dir ? b_ih_b : b_ih_f;
    const float* Bhh = dir ? b_hh_b : b_hh_f;

    // weights in WMMA A(16x4 f32) layout: lane = gate row, K slice across VGPR pair / lane halves
    v2f Aih[3][4], Ahh[3][4];
    v8f Cih[3];   // bias tiles in C layout; r,z gates get b_ih + b_hh folded together
    v8f Chh_n;    // n gate keeps b_hh separate (it is multiplied by r later)
#pragma unroll
    for (int g = 0; g < 3; ++g) {
#pragma unroll
      for (int j = 0; j < 4; ++j) {
        int col = 4 * j + (hi ? 2 : 0);
        const float* pi = Wih + (g * 16 + m16) * 16 + col;
        const float* ph = Whh + (g * 16 + m16) * 16 + col;
        Aih[g][j] = mkv2(pi[0], pi[1]);
        Ahh[g][j] = mkv2(ph[0], ph[1]);
      }
      v8f ci;
#pragma unroll
      for (int r = 0; r < 8; ++r) {
        int c = g * 16 + r + (hi ? 8 : 0);
        ci[r] = Bih[c] + ((g < 2) ? Bhh[c] : 0.0f);
      }
      Cih[g] = ci;
    }
#pragma unroll
    for (int r = 0; r < 8; ++r) Chh_n[r] = Bhh[2 * 16 + r + (hi ? 8 : 0)];

    v8f h = vzero;  // h^T tile: [hid, seq] in D layout

    for (int step = 0; step < GL; ++step) {
      const int p = dir ? (GL - 1 - step) : step;

      // bucket for (s = m16, position p) from packed registers
      unsigned buck = ((p < 10) ? (bp0 >> (3 * p)) : (bp1 >> (3 * (p - 10)))) & 7u;

      // x_t^T in B(4x16) slice layout, gathered straight from the embedding table:
      // VGPR 2j+rs holds rows 4j+rs (lo lanes) / 4j+rs+2 (hi lanes)
      const float* eb = elds + buck * 16 + (hi ? 2 : 0);
      float xB[8];
      xB[0] = eb[0];  xB[1] = eb[1];
      xB[2] = eb[4];  xB[3] = eb[5];
      xB[4] = eb[8];  xB[5] = eb[9];
      xB[6] = eb[12]; xB[7] = eb[13];

      // x_gates = W_ih * x^T + (b_ih [+ b_hh for r,z])   (3 gate tiles x 4 chained K=4 WMMAs)
      v8f Xg[3];
#pragma unroll
      for (int g = 0; g < 3; ++g) {
        v8f c = Cih[g];
#pragma unroll
        for (int j = 0; j < 4; ++j)
          c = wmma_k4(Aih[g][j], mkv2(xB[2 * j], xB[2 * j + 1]), c);
        Xg[g] = c;
      }

      // h (D layout) -> B slices: only half-wave swaps needed
      float sw[8];
#pragma unroll
      for (int r = 0; r < 8; ++r) sw[r] = __shfl_xor((float)h[r], 16, 32);
      float hB[8];
      hB[0] = hi ? sw[2]       : (float)h[0];
      hB[1] = hi ? sw[3]       : (float)h[1];
      hB[2] = hi ? sw[6]       : (float)h[4];
      hB[3] = hi ? sw[7]       : (float)h[5];
      hB[4] = hi ? (float)h[2] : sw[0];
      hB[5] = hi ? (float)h[3] : sw[1];
      hB[6] = hi ? (float)h[6] : sw[4];
      hB[7] = hi ? (float)h[7] : sw[5];

      // hg = W_hh * h^T  (+ b_hh only for the n gate)
      v8f Hg[3];
#pragma unroll
      for (int g = 0; g < 3; ++g) {
        v8f c = (g == 2) ? Chh_n : vzero;
#pragma unroll
        for (int j = 0; j < 4; ++j)
          c = wmma_k4(Ahh[g][j], mkv2(hB[2 * j], hB[2 * j + 1]), c);
        Hg[g] = c;
      }

      // GRU update (PyTorch gate order r,z,n)
#pragma unroll
      for (int r = 0; r < 8; ++r) {
        float rg = fsigmoid(Xg[0][r] + Hg[0][r]);
        float zg = fsigmoid(Xg[1][r] + Hg[1][r]);
        float ng = ftanh_(Xg[2][r] + rg * Hg[2][r]);
        h[r] = (1.0f - zg) * ng + zg * h[r];
      }

      if (dir == 0) {
        // stash forward output tile for position p
#pragma unroll
        for (int r = 0; r < 8; ++r)
          flds[(p * 16 + r + (hi ? 8 : 0)) * 16 + m16] = h[r];
      } else {
        // fuse: read forward tile, layernorm over 32 channels, staged coalesced store
        v8f hf;
#pragma unroll
        for (int r = 0; r < 8; ++r)
          hf[r] = flds[(p * 16 + r + (hi ? 8 : 0)) * 16 + m16];

        float sum = 0.f, sq = 0.f;
#pragma unroll
        for (int r = 0; r < 8; ++r) {
          sum += hf[r] + h[r];
          sq  += hf[r] * hf[r] + (float)h[r] * (float)h[r];
        }
        sum += __shfl_xor(sum, 16, 32);   // lane l and l^16 hold the same sequence
        sq  += __shfl_xor(sq, 16, 32);
        float mu  = sum * (1.0f / 32.0f);
        float var = sq * (1.0f / 32.0f) - mu * mu;
        float inv = rsqrtf(var + 1e-12f);

#pragma unroll
        for (int r = 0; r < 8; ++r) {
          int cf = r + (hi ? 8 : 0);
          ystage[m16 * 32 + cf]      = (hf[r] - mu) * inv * gamF[r] + betF[r];
          ystage[m16 * 32 + cf + 16] = ((float)h[r] - mu) * inv * gamB[r] + betB[r];
        }
        // 16 seqs x 32 ch = 128 b128 stores, 128B contiguous per sequence-position
#pragma unroll
        for (int it = 0; it < 4; ++it) {
          int idx = it * 32 + lane;
          int s = idx >> 3, c4 = idx & 7;
          long n = nbase + s;
          if (n < (long)n_seq) {
            v4f y = *(const v4f*)(ystage + s * 32 + c4 * 4);
            *(v4f*)(out + (n * GL + p) * 32 + c4 * 4) = y;
          }
        }
      }
    }
    __syncthreads();  // single-wave WG: lowers to S_NOP, kept for clarity
  }
}

extern "C" void kernel_launch(void* const* d_in, const int* in_sizes, int n_in,
                              void* d_out, int out_size, void* d_ws, size_t ws_size,
                              hipStream_t stream) {
  const int*   ids    = (const int*)d_in[0];
  const float* emb    = (const float*)d_in[1];
  const float* w_ih_f = (const float*)d_in[2];
  const float* w_hh_f = (const float*)d_in[3];
  const float* b_ih_f = (const float*)d_in[4];
  const float* b_hh_f = (const float*)d_in[5];
  const float* w_ih_b = (const float*)d_in[6];
  const float* w_hh_b = (const float*)d_in[7];
  const float* b_ih_b = (const float*)d_in[8];
  const float* b_hh_b = (const float*)d_in[9];
  const float* gamma  = (const float*)d_in[10];
  const float* beta   = (const float*)d_in[11];
  float* out = (float*)d_out;

  int n_seq  = in_sizes[0] / GL;            // B*T sequences
  int ntiles = (n_seq + 15) / 16;           // one wave handles 16 sequences

  bigru_ln_kernel<<<dim3(ntiles), dim3(32), 0, stream>>>(
      ids, emb, w_ih_f, w_hh_f, b_ih_f, b_hh_f,
      w_ih_b, w_hh_b, b_ih_b, b_hh_b, gamma, beta, out, n_seq);
}